// SegLoss_31078383354717
// MI455X (gfx1250) — compile-verified
//
#include <hip/hip_runtime.h>
#include <hip/hip_bf16.h>

// ---------------------------------------------------------------------------
// SegLoss (BCE-with-logits + Dice) for B=16, H=W=1024, N=2000 points, r=2.
// Memory-bound streaming reduction: 67MB logits + 2MB gt bitmask, ~3us at
// 23.3 TB/s HBM. Inner loop strength-reduced to ~16 VALU (3 TRANS) per
// element (single exp, v_rcp instead of IEEE divide, v_log instead of log1p
// libcall) so the kernel stays HBM-bound, not VALU-bound.
// WMMA (V_WMMA_F32_16X16X4_F32) performs the wave-level reduction of the five
// accumulators (A*ones trick: D[m][n] = v_m + v_{m+16}).
// Deterministic: no float atomics; fixed-order partial-sum reductions.
// ---------------------------------------------------------------------------

typedef __attribute__((ext_vector_type(2))) float v2f;
typedef __attribute__((ext_vector_type(8))) float v8f;

#define HDIM 1024
#define WDIM 1024
#define BATCH 16
#define HW (HDIM * WDIM)
#define NPTS 2000
#define THREADS 256
#define BLOCKS_PER_IMG 128
#define PIX_PER_BLOCK (HW / BLOCKS_PER_IMG)            // 8192
#define F4_PER_THREAD (PIX_PER_BLOCK / (THREADS * 4))  // 8
#define NBLK (BATCH * BLOCKS_PER_IMG)                  // 2048
#define MASK_WORDS (BATCH * HW / 32)                   // 524288 u32 = 2 MB
#define SMOOTHF 1e-6f

// Full-wave (32 lane) sum via f32 WMMA. EXEC must be all ones at call site.
// A layout (16x4 f32): lanes 0-15 vgpr0 -> A[m][0], lanes 16-31 vgpr0 -> A[m][2].
// With a={v,0}, B=ones: D[m][n] = v_m + v_{m+16}. Each lane holds 8 rows of D
// (lanes<16: rows 0-7, lanes>=16: rows 8-15); sum those, then xor-shuffle by 16.
__device__ __forceinline__ float wave_sum_wmma(float v) {
  v2f a; a[0] = v;    a[1] = 0.0f;
  v2f b; b[0] = 1.0f; b[1] = 1.0f;
  v8f c = {};
  c = __builtin_amdgcn_wmma_f32_16x16x4_f32(
      /*neg_a=*/false, a, /*neg_b=*/false, b,
      /*c_mod=*/(short)0, c, /*reuse_a=*/false, /*reuse_b=*/false);
  float s = ((c[0] + c[1]) + (c[2] + c[3])) + ((c[4] + c[5]) + (c[6] + c[7]));
  s += __shfl_xor(s, 16, 32);
  return s;
}

__global__ void zero_mask_kernel(unsigned* __restrict__ mask) {
  int i = blockIdx.x * blockDim.x + threadIdx.x;
  if (i < MASK_WORDS) mask[i] = 0u;
}

// One thread per point: rasterize radius-2 disk (13 pixels) as bits.
__global__ void raster_kernel(const int* __restrict__ pts,
                              unsigned* __restrict__ mask) {
  int tid = blockIdx.x * blockDim.x + threadIdx.x;
  if (tid >= BATCH * NPTS) return;
  int b  = tid / NPTS;
  int px = pts[tid * 2 + 0];
  int py = pts[tid * 2 + 1];
  int cx = px >> 2;  // floor-div by STRIDE=4 (inputs non-negative)
  int cy = py >> 2;
  if (cx < 0 || cx >= WDIM || cy < 0 || cy >= HDIM) return;
  const int dys[13] = {-2,-1,-1,-1, 0, 0, 0, 0, 0, 1, 1, 1, 2};
  const int dxs[13] = { 0,-1, 0, 1,-2,-1, 0, 1, 2,-1, 0, 1, 0};
#pragma unroll
  for (int k = 0; k < 13; ++k) {
    int yy = cy + dys[k], xx = cx + dxs[k];
    if (yy >= 0 && yy < HDIM && xx >= 0 && xx < WDIM) {
      int idx = b * HW + yy * WDIM + xx;
      atomicOr(&mask[idx >> 5], 1u << (idx & 31));
    }
  }
}

// Streaming pass: each block owns a contiguous 8192-pixel chunk of one image.
// Per-element (g = gt bit):
//   e  = exp(-|x|)
//   sp = softplus(x) = max(x,0) + log(1+e)        (softplus(-x) = sp - x)
//   pr = sigmoid(x)  = (x>=0 ? 1 : e) * rcp(1+e)
// Accumulate Σsp, Σg·sp, Σg·x, Σpr, Σg·pr, Σg; derive
//   a0 = Σ(1-g)·softplus(x) = Σsp − Σg·sp
//   a1 = Σg·softplus(-x)    = Σg·sp − Σg·x
__global__ void seg_main_kernel(const float* __restrict__ logits,
                                const unsigned* __restrict__ mask,
                                float* __restrict__ part) {
  int blk = blockIdx.x;
  int b = blk / BLOCKS_PER_IMG;
  int j = blk % BLOCKS_PER_IMG;
  long base = (long)b * HW + (long)j * PIX_PER_BLOCK;
  const float4* lg4 = reinterpret_cast<const float4*>(logits + base);
  int t = threadIdx.x;

  float spsum = 0.f, spg = 0.f, xg = 0.f, p = 0.f, ig = 0.f, gc = 0.f;
#pragma unroll
  for (int k = 0; k < F4_PER_THREAD; ++k) {
    int f4 = k * THREADS + t;               // coalesced b128 loads
    float4 x = lg4[f4];
    long gidx = base + (long)f4 * 4;
    unsigned mb = mask[(unsigned)(gidx >> 5)] >> ((int)gidx & 31);
    float xs[4] = {x.x, x.y, x.z, x.w};
#pragma unroll
    for (int e4 = 0; e4 < 4; ++e4) {
      float xv = xs[e4];
      float ax = fabsf(xv);
      float e  = __expf(-ax);                       // v_exp_f32
      float tn = 1.f + e;
      float sp = fmaxf(xv, 0.f) + __logf(tn);      // v_log_f32
      float r  = __builtin_amdgcn_rcpf(tn);        // v_rcp_f32
      float pr = (xv >= 0.f ? 1.f : e) * r;        // sigmoid(x)
      float g  = (float)((mb >> e4) & 1u);
      spsum += sp;
      p     += pr;
      spg = fmaf(g, sp, spg);
      xg  = fmaf(g, xv, xg);
      ig  = fmaf(g, pr, ig);
      gc  += g;
    }
  }
  float a0 = spsum - spg;   // Σ(1-g)*softplus(x)
  float a1 = spg - xg;      // Σ g*softplus(-x)

  // Wave-level reduction on the CDNA5 matrix path (all lanes active here).
  float w0 = wave_sum_wmma(a0);
  float w1 = wave_sum_wmma(a1);
  float wp = wave_sum_wmma(p);
  float wi = wave_sum_wmma(ig);
  float wg = wave_sum_wmma(gc);

  __shared__ float red[THREADS / 32][5];
  int wave = t >> 5, lane = t & 31;
  if (lane == 0) {
    red[wave][0] = w0; red[wave][1] = w1; red[wave][2] = wp;
    red[wave][3] = wi; red[wave][4] = wg;
  }
  __syncthreads();
  if (t == 0) {
    float s0 = 0.f, s1 = 0.f, s2 = 0.f, s3 = 0.f, s4 = 0.f;
    for (int w = 0; w < THREADS / 32; ++w) {  // fixed order -> deterministic
      s0 += red[w][0]; s1 += red[w][1]; s2 += red[w][2];
      s3 += red[w][3]; s4 += red[w][4];
    }
    part[0 * NBLK + blk] = s0;
    part[1 * NBLK + blk] = s1;
    part[2 * NBLK + blk] = s2;
    part[3 * NBLK + blk] = s3;
    part[4 * NBLK + blk] = s4;
  }
}

// Single wave: deterministic fixed-order reduction of 2048 partials + scalar.
__global__ void finalize_kernel(const float* __restrict__ part,
                                float* __restrict__ out) {
  int t = threadIdx.x;  // 32 lanes
  float s0 = 0.f, s1 = 0.f;
  for (int k = 0; k < NBLK / 32; ++k) {
    int idx = t * (NBLK / 32) + k;
    s0 += part[0 * NBLK + idx];
    s1 += part[1 * NBLK + idx];
  }
  for (int off = 16; off; off >>= 1) {
    s0 += __shfl_xor(s0, off, 32);
    s1 += __shfl_xor(s1, off, 32);
  }
  float frac = 0.f, gtot = 0.f;
  if (t < BATCH) {
    float P = 0.f, I = 0.f, G = 0.f;
    for (int j = 0; j < BLOCKS_PER_IMG; ++j) {
      int idx = t * BLOCKS_PER_IMG + j;
      P += part[2 * NBLK + idx];
      I += part[3 * NBLK + idx];
      G += part[4 * NBLK + idx];
    }
    frac = (2.f * I + SMOOTHF) / (P + G + SMOOTHF);
    gtot = G;
  }
  for (int off = 16; off; off >>= 1) {
    frac += __shfl_xor(frac, off, 32);
    gtot += __shfl_xor(gtot, off, 32);
  }
  if (t == 0) {
    float total = (float)BATCH * (float)HW;
    float pos = gtot;
    float neg = total - pos;
    float pw  = neg / (pos + SMOOTHF);
    float bce = (pw * s1 + s0) / total;
    float dice = 1.f - frac / (float)BATCH;
    out[0] = 0.5f * bce + 0.5f * dice;
  }
}

extern "C" void kernel_launch(void* const* d_in, const int* in_sizes, int n_in,
                              void* d_out, int out_size, void* d_ws, size_t ws_size,
                              hipStream_t stream) {
  const float* logits = (const float*)d_in[0];   // [16,1,1024,1024] f32
  const int*   pts    = (const int*)d_in[1];     // [16,2000,2] i32
  unsigned* mask = (unsigned*)d_ws;                                   // 2 MB
  float*    part = (float*)((char*)d_ws + (size_t)MASK_WORDS * 4u);   // 5*2048 f32

  zero_mask_kernel<<<MASK_WORDS / 256, 256, 0, stream>>>(mask);
  raster_kernel<<<(BATCH * NPTS + 255) / 256, 256, 0, stream>>>(pts, mask);
  seg_main_kernel<<<NBLK, THREADS, 0, stream>>>(logits, mask, part);
  finalize_kernel<<<1, 32, 0, stream>>>(part, (float*)d_out);
}